// BertCrf4Role_41231686041659
// MI455X (gfx1250) — compile-verified
//
#include <hip/hip_runtime.h>
#include <hip/hip_bf16.h>

// ---------------------------------------------------------------------------
// BERT-CRF: emissions GEMM (f32 WMMA 16x16x4, W staged to LDS via TDM) +
// CRF forward/Viterbi scan + backtrace.  B=128, T=256, H=768, K=64.
// ---------------------------------------------------------------------------

typedef __attribute__((ext_vector_type(2))) float v2f;
typedef __attribute__((ext_vector_type(8))) float v8f;
typedef __attribute__((ext_vector_type(4))) unsigned int u32x4;
typedef __attribute__((ext_vector_type(8))) int i32x8;
typedef __attribute__((ext_vector_type(4))) int i32x4;

#define BB 128
#define TT 256
#define HH 768
#define KK 64

// ---------------------------------------------------------------------------
// Kernel 1: emissions[b,t,k] = hidden[b,t,:] @ W[:,k] + bias[k]
// M = B*T = 32768 rows. Grid: 256 blocks x 256 threads (8 waves).
// W (768x64 f32 = 192 KB) is DMA'd to LDS once per block by the Tensor Data
// Mover (tensor_load_to_lds, TENSORcnt), then B-fragments come from LDS.
// A-fragments stream from global (the 100 MB HBM-bound side).
// ---------------------------------------------------------------------------
__global__ __launch_bounds__(256) void emis_gemm_wmma(
    const float* __restrict__ hidden,   // [32768, 768]
    const float* __restrict__ W,        // [768, 64]
    const float* __restrict__ bias,     // [64]
    float* __restrict__ emis)           // [32768, 64]
{
    extern __shared__ float w_lds[];                   // 768*64 floats = 192 KB

    const int lane = threadIdx.x & 31;
    const int wave = threadIdx.x >> 5;                 // 0..7
    const int row_base = blockIdx.x * 128 + wave * 16; // 16 rows per wave
    const int h = lane >> 4;                           // 0 or 1
    const int r = lane & 15;

    // --- TDM: stage W into LDS (one wave issues; 2D tile 64 x 768, 4B elems)
    if (wave == 0) {
        const unsigned lds_base = (unsigned)(uintptr_t)&w_lds[0];
        const unsigned long long ga = (unsigned long long)(uintptr_t)W;

        u32x4 g0;
        g0[0] = 1u;                                    // count=1 (valid), no gather
        g0[1] = lds_base;                              // D#.lds_addr
        g0[2] = (unsigned)(ga & 0xFFFFFFFFu);          // global_addr[31:0]
        g0[3] = (unsigned)((ga >> 32) & 0x01FFFFFFu)   // global_addr[56:32]
                | (2u << 30);                          // type = 2 ("image")

        i32x8 g1;
        g1[0] = 0x00020000;          // data_size=2 (4B), workgroup_mask=0
        g1[1] = (int)(64u << 16);    // tensor_dim0 = 64  (bits 63:48)
        g1[2] = (int)(768u << 16);   // tensor_dim1 = 768 (bits 95:80)
        g1[3] = (int)(64u << 16);    // tile_dim0 = 64    (bits 127:112)
        g1[4] = 768;                 // tile_dim1 = 768   (bits 143:128)
        g1[5] = 64;                  // tensor_dim0_stride = 64 elems
        g1[6] = 0;
        g1[7] = 0;

        i32x4 gz  = {0, 0, 0, 0};    // 2D tensor: groups 2/3 unused
        i32x8 gz8 = {0, 0, 0, 0, 0, 0, 0, 0};

        __builtin_amdgcn_tensor_load_to_lds(g0, g1, gz, gz, gz8, 0);
        __builtin_amdgcn_s_wait_tensorcnt(0);
    }
    __syncthreads();

    const float* __restrict__ arow = hidden + (size_t)(row_base + r) * HH;

    v8f acc0 = {}, acc1 = {}, acc2 = {}, acc3 = {};

    for (int k0 = 0; k0 < HH; k0 += 4) {
        const int kk = k0 + 2 * h;
        // prefetch A data a little ahead (global_prefetch_b8)
        __builtin_prefetch(arow + kk + 64, 0, 0);

        v2f a = *(const v2f*)(arow + kk);              // K=kk, kk+1 of this row

        const float* __restrict__ w0 = w_lds + kk * KK;    // K=kk row (LDS)
        const float* __restrict__ w1 = w0 + KK;            // K=kk+1 row (LDS)
        v2f b0, b1, b2, b3;
        b0.x = w0[r     ]; b0.y = w1[r     ];
        b1.x = w0[r + 16]; b1.y = w1[r + 16];
        b2.x = w0[r + 32]; b2.y = w1[r + 32];
        b3.x = w0[r + 48]; b3.y = w1[r + 48];

        acc0 = __builtin_amdgcn_wmma_f32_16x16x4_f32(false, a, false, b0,
                                                     (short)0, acc0, false, false);
        acc1 = __builtin_amdgcn_wmma_f32_16x16x4_f32(false, a, false, b1,
                                                     (short)0, acc1, false, false);
        acc2 = __builtin_amdgcn_wmma_f32_16x16x4_f32(false, a, false, b2,
                                                     (short)0, acc2, false, false);
        acc3 = __builtin_amdgcn_wmma_f32_16x16x4_f32(false, a, false, b3,
                                                     (short)0, acc3, false, false);
    }

    // Epilogue: D layout = VGPR i holds M = i (lanes 0-15) / M = i+8 (lanes
    // 16-31), N = lane&15.  Add bias, store.
    const float bb0 = bias[r];
    const float bb1 = bias[r + 16];
    const float bb2 = bias[r + 32];
    const float bb3 = bias[r + 48];
    #pragma unroll
    for (int i = 0; i < 8; ++i) {
        const int orow = row_base + h * 8 + i;
        float* __restrict__ o = emis + (size_t)orow * KK;
        o[r     ] = acc0[i] + bb0;
        o[r + 16] = acc1[i] + bb1;
        o[r + 32] = acc2[i] + bb2;
        o[r + 48] = acc3[i] + bb3;
    }
}

// ---------------------------------------------------------------------------
// Kernel 2: per-sequence CRF scan.  Grid: 128 blocks (one per b), 256 threads.
// Thread layout: j = tid>>2 (state 0..63), sub = tid&3 (16 predecessor states
// each); width-4 __shfl_xor reductions combine the partial logsumexp /
// max-argmax.  transitions staged in LDS with pitch 65 (conflict-free).
// ---------------------------------------------------------------------------
__global__ __launch_bounds__(256) void crf_scan(
    const float* __restrict__ emis,       // [128, 256, 64]
    const int*   __restrict__ masks,      // [128, 256]
    const int*   __restrict__ target,     // [128, 256]
    const float* __restrict__ trans,      // [64, 64]
    float*       __restrict__ out,        // d_out: [32768 path][128 ll]
    unsigned char* __restrict__ bp,       // [128, 256, 64]
    int*         __restrict__ last_tag,   // [128]
    int*         __restrict__ seqlen_out) // [128]
{
    __shared__ float trans_sh[KK * 65];
    __shared__ float alpha_sh[KK];
    __shared__ float v_sh[KK];
    __shared__ float redf[256];
    __shared__ int   redi[256];
    __shared__ int   s_seqlen;
    __shared__ float s_lognorm;

    const int b   = blockIdx.x;
    const int tid = threadIdx.x;
    const float NEG_INF = -__builtin_inff();

    // stage transitions (pitch 65)
    #pragma unroll
    for (int i = 0; i < (KK * KK) / 256; ++i) {
        const int idx = tid + i * 256;
        trans_sh[(idx >> 6) * 65 + (idx & 63)] = trans[idx];
    }

    // seq_len = sum(masks[b, :])
    redi[tid] = masks[b * TT + tid];
    __syncthreads();
    for (int st = 128; st > 0; st >>= 1) {
        if (tid < st) redi[tid] += redi[tid + st];
        __syncthreads();
    }
    if (tid == 0) s_seqlen = redi[0];

    // alpha0 / v0 = emissions[b, 0, :]
    if (tid < KK) {
        const float e0 = emis[(size_t)b * TT * KK + tid];
        alpha_sh[tid] = e0;
        v_sh[tid]     = e0;
    }
    __syncthreads();

    const int j   = tid >> 2;
    const int sub = tid & 3;
    const int k_lo = sub * 16;

    for (int t = 1; t < TT; ++t) {
        // partial logsumexp + Viterbi max over k in [k_lo, k_lo+16)
        float m = NEG_INF, s = 0.0f;
        float mv = NEG_INF;
        int   arg = k_lo;
        #pragma unroll
        for (int kk = 0; kk < 16; ++kk) {
            const int k = k_lo + kk;
            const float tr = trans_sh[k * 65 + j];
            const float x = alpha_sh[k] + tr;
            if (x > m) { s = s * __expf(m - x) + 1.0f; m = x; }
            else       { s += __expf(x - m); }
            const float y = v_sh[k] + tr;
            if (y > mv) { mv = y; arg = k; }
        }
        // combine 4 sub-lanes (adjacent lanes, wave32 shuffle, width 4)
        #pragma unroll
        for (int off = 1; off < 4; off <<= 1) {
            const float m2  = __shfl_xor(m,  off, 4);
            const float s2  = __shfl_xor(s,  off, 4);
            const float nm  = fmaxf(m, m2);
            s = s * __expf(m - nm) + s2 * __expf(m2 - nm);
            m = nm;
            const float mv2 = __shfl_xor(mv, off, 4);
            const int   a2  = __shfl_xor(arg, off, 4);
            if (mv2 > mv || (mv2 == mv && a2 < arg)) { mv = mv2; arg = a2; }
        }

        const float e  = emis[((size_t)b * TT + t) * KK + j];
        const float na = m + __logf(s) + e;
        const float nv = mv + e;
        const bool valid = (t < s_seqlen);

        __syncthreads();   // all reads of alpha_sh/v_sh done
        if (sub == 0) {
            if (valid) { alpha_sh[j] = na; v_sh[j] = nv; }
            bp[((size_t)b * TT + t) * KK + j] =
                (unsigned char)(valid ? arg : j);
        }
        __syncthreads();
    }

    // log_norm = logsumexp(alpha), best/last_tag = max/argmax(v)  (tiny: serial)
    if (tid == 0) {
        float m = NEG_INF, s = 0.0f;
        float bm = NEG_INF; int ba = 0;
        for (int k = 0; k < KK; ++k) {
            const float x = alpha_sh[k];
            if (x > m) { s = s * __expf(m - x) + 1.0f; m = x; }
            else       { s += __expf(x - m); }
            if (v_sh[k] > bm) { bm = v_sh[k]; ba = k; }
        }
        s_lognorm = m + __logf(s);
        last_tag[b]   = ba;
        seqlen_out[b] = s_seqlen;
    }
    __syncthreads();

    // sequence score (unary + pairwise), one t per thread
    {
        const int t = tid;
        float contrib = 0.0f;
        if (t < s_seqlen) {
            const int tg = target[b * TT + t];
            contrib = emis[((size_t)b * TT + t) * KK + tg];
            if (t >= 1) {
                const int tp = target[b * TT + t - 1];
                contrib += trans_sh[tp * 65 + tg];
            }
        }
        redf[tid] = contrib;
        __syncthreads();
        for (int st = 128; st > 0; st >>= 1) {
            if (tid < st) redf[tid] += redf[tid + st];
            __syncthreads();
        }
        if (tid == 0) out[(size_t)BB * TT + b] = redf[0] - s_lognorm;
    }
}

// ---------------------------------------------------------------------------
// Kernel 3: backtrace.  One thread per sequence.
// ---------------------------------------------------------------------------
__global__ void crf_backtrace(
    const unsigned char* __restrict__ bp,
    const int* __restrict__ last_tag,
    const int* __restrict__ seqlen,
    float* __restrict__ out)             // path as floats, [128, 256]
{
    const int b = threadIdx.x;
    if (b >= BB) return;
    const int sl = seqlen[b];
    int tag = last_tag[b];
    out[b * TT + (TT - 1)] = ((TT - 1) < sl) ? (float)tag : 0.0f;
    for (int t = TT - 1; t >= 1; --t) {
        tag = bp[((size_t)b * TT + t) * KK + tag];
        out[b * TT + (t - 1)] = ((t - 1) < sl) ? (float)tag : 0.0f;
    }
}

// ---------------------------------------------------------------------------
extern "C" void kernel_launch(void* const* d_in, const int* in_sizes, int n_in,
                              void* d_out, int out_size, void* d_ws, size_t ws_size,
                              hipStream_t stream) {
    const float* hidden = (const float*)d_in[0];
    const int*   masks  = (const int*)d_in[1];
    const int*   target = (const int*)d_in[2];
    const float* W      = (const float*)d_in[3];
    const float* bias   = (const float*)d_in[4];
    const float* trans  = (const float*)d_in[5];
    float* out = (float*)d_out;

    char* ws = (char*)d_ws;
    float*         emis     = (float*)ws;                      // 8 MB
    unsigned char* bp       = (unsigned char*)(ws + 8388608);  // 2 MB
    int*           last_tag = (int*)(ws + 8388608 + 2097152);  // 512 B
    int*           seqlen   = last_tag + BB;                   // 512 B

    emis_gemm_wmma<<<256, 256, HH * KK * sizeof(float), stream>>>(
        hidden, W, bias, emis);
    crf_scan<<<BB, 256, 0, stream>>>(emis, masks, target, trans,
                                     out, bp, last_tag, seqlen);
    crf_backtrace<<<1, BB, 0, stream>>>(bp, last_tag, seqlen, out);
}